// GCN_63677185131176
// MI455X (gfx1250) — compile-verified
//
#include <hip/hip_runtime.h>
#include <hip/hip_bf16.h>

#define HID  128
#define NGR  512
#define FINF 7
#define OUTF 2

typedef float v2f __attribute__((ext_vector_type(2)));
typedef float v8f __attribute__((ext_vector_type(8)));

// ---------------- degree / normalization ----------------

__global__ void init_deg_kernel(float* __restrict__ deg, int n) {
    int i = blockIdx.x * blockDim.x + threadIdx.x;
    if (i < n) deg[i] = 1.0f;                 // self-loop contribution
}

__global__ void edge_deg_kernel(const int* __restrict__ dst,
                                float* __restrict__ deg, int e) {
    int i = blockIdx.x * blockDim.x + threadIdx.x;
    if (i < e) atomicAdd(&deg[dst[i]], 1.0f);
}

__global__ void dinv_kernel(float* __restrict__ deg, int n) {
    int i = blockIdx.x * blockDim.x + threadIdx.x;
    if (i < n) {
        float d = deg[i];
        deg[i] = (d > 0.0f) ? rsqrtf(d) : 0.0f;   // in-place deg -> dinv
    }
}

__global__ void norm_kernel(const int* __restrict__ src, const int* __restrict__ dst,
                            const float* __restrict__ dinv,
                            float* __restrict__ norm, int e) {
    int i = blockIdx.x * blockDim.x + threadIdx.x;
    if (i < e) norm[i] = dinv[src[i]] * dinv[dst[i]];
}

// ---------------- fp32 WMMA GEMM:  Out[rows x 128] = (relu?)H[rows x K] @ W[K x 128] ----------------
// One wave -> 16x16 output tile via V_WMMA_F32_16X16X4_F32.
// Block = 256 threads = 8 waves = all 128 output cols of one 16-row block.
// A tile staged once per block in LDS (relu + zero-pad fused), read back as ds_load_b64 fragments.
// TAIL=false path (the full grid) is completely branch/guard free.

template<int K, bool RELU, bool TAIL>
__global__ __launch_bounds__(256)
void gemm_wmma_kernel(const float* __restrict__ H,
                      const float* __restrict__ W,
                      float* __restrict__ Out,
                      int nrows, int rblockBase) {
    constexpr int KP  = (K + 7) & ~7;     // K padded to multiple of 8 (8 or 128)
    constexpr int LDA = KP + 4;           // LDS row stride: stride%64==4 -> conflict-free fragment reads
    __shared__ float As[16 * LDA];

    const int tid  = threadIdx.x;
    const int lane = tid & 31;
    const int wave = tid >> 5;
    const int half = lane >> 4;           // 0: lanes 0-15, 1: lanes 16-31
    const int l15  = lane & 15;
    const int r0   = (rblockBase + blockIdx.x) * 16;   // row block base
    const int n0   = wave * 16;                        // column tile base

    // ---- cooperative A-tile load: 16 x KP floats, coalesced along K ----
    #pragma unroll
    for (int i = tid; i < 16 * KP; i += 256) {
        const int row = i / KP;           // KP is a power of two -> shifts
        const int col = i % KP;
        float v = 0.0f;
        if (col < K) {                    // compile-time folded when K == KP
            int m = r0 + row;
            if (TAIL) m = (m < nrows) ? m : (nrows - 1);
            v = H[(long)m * K + col];
            if (RELU) v = fmaxf(v, 0.0f);
        }
        As[row * LDA + col] = v;
    }
    __syncthreads();

    v8f c = {0.f, 0.f, 0.f, 0.f, 0.f, 0.f, 0.f, 0.f};

    #pragma unroll 8
    for (int k0 = 0; k0 < KP; k0 += 4) {
        const int kk = k0 + 2 * half;     // A frag: lanes 0-15 hold K=kk,kk+1; lanes 16-31 K=kk+2,kk+3
        const v2f a = *(const v2f*)(&As[l15 * LDA + kk]);   // 8B-aligned -> ds_load_b64

        // B frag (4x16): rows kk, kk+1 striped across lanes
        float bx, by;
        if constexpr (KP == K) {
            bx = W[kk * HID + n0 + l15];
            by = W[(kk + 1) * HID + n0 + l15];
        } else {
            // unconditional clamped loads + cndmask zeroing: no exec-mask branches
            const int kc0 = (kk     < K) ? kk     : 0;
            const int kc1 = (kk + 1 < K) ? kk + 1 : 0;
            bx = W[kc0 * HID + n0 + l15];
            by = W[kc1 * HID + n0 + l15];
            bx = (kk     < K) ? bx : 0.0f;
            by = (kk + 1 < K) ? by : 0.0f;
        }
        v2f b = {bx, by};

        c = __builtin_amdgcn_wmma_f32_16x16x4_f32(
                /*neg_a=*/false, a, /*neg_b=*/false, b,
                /*c_mod=*/(short)0, c, /*reuse_a=*/false, /*reuse_b=*/false);
    }

    // D tile: VGPR r holds row (r0 + r + 8*half), col n0 + l15.
    // Base pointer hoisted -> 8 stores with immediate offsets (r*HID*4 bytes).
    float* o = Out + (long)(r0 + 8 * half) * HID + n0 + l15;
    if constexpr (!TAIL) {
        #pragma unroll
        for (int r = 0; r < 8; ++r)
            o[r * HID] = c[r];
    } else {
        const int rowBase = r0 + 8 * half;
        #pragma unroll
        for (int r = 0; r < 8; ++r)
            if (rowBase + r < nrows)
                o[r * HID] = c[r];
    }
}

// ---------------- agg init: bias + self-loop term ----------------

__global__ void init_agg_kernel(const float* __restrict__ hW,
                                const float* __restrict__ dinv,
                                const float* __restrict__ bias,
                                float* __restrict__ agg, long total) {
    long i = (long)blockIdx.x * blockDim.x + threadIdx.x;
    if (i < total) {
        int node = (int)(i >> 7);
        int ch   = (int)(i & (HID - 1));
        float di = dinv[node];
        agg[i] = bias[ch] + di * di * hW[i];
    }
}

// ---------------- edge scatter: agg[dst] += norm * hW[src]  (float4 per thread, 32 threads/edge) ----

__global__ void edge_scatter_kernel(const int* __restrict__ src,
                                    const int* __restrict__ dst,
                                    const float* __restrict__ norm,
                                    const float* __restrict__ hW,
                                    float* __restrict__ agg, long total) {
    long idx = (long)blockIdx.x * blockDim.x + threadIdx.x;
    if (idx < total) {
        // e is identical for all 32 lanes of the wave -> pin to SGPR so the
        // edge metadata comes in as scalar loads instead of broadcast VMEM.
        int e  = __builtin_amdgcn_readfirstlane((int)(idx >> 5));
        int c4 = ((int)idx & 31) * 4;      // 32 threads * 4 ch = 128 channels
        int s = src[e];
        int d = dst[e];
        float w = norm[e];
        const float4 v = *(const float4*)(hW + (long)s * HID + c4);
        float* o = agg + (long)d * HID + c4;
        atomicAdd(o + 0, w * v.x);
        atomicAdd(o + 1, w * v.y);
        atomicAdd(o + 2, w * v.z);
        atomicAdd(o + 3, w * v.w);
    }
}

// ---------------- pooling ----------------

__global__ void pool_zero_kernel(float* __restrict__ sums, float* __restrict__ cnt) {
    int i = blockIdx.x * blockDim.x + threadIdx.x;
    if (i < NGR * HID) sums[i] = 0.0f;
    if (i < NGR) cnt[i] = 0.0f;
}

__global__ void pool_scatter_kernel(const float* __restrict__ h,
                                    const int* __restrict__ batch,
                                    float* __restrict__ sums,
                                    float* __restrict__ cnt, long total) {
    long i = (long)blockIdx.x * blockDim.x + threadIdx.x;
    if (i < total) {
        int node = (int)(i >> 7);
        int ch   = (int)(i & (HID - 1));
        int g = batch[node];
        atomicAdd(&sums[(long)g * HID + ch], h[i]);
        if (ch == 0) atomicAdd(&cnt[g], 1.0f);
    }
}

__global__ void head_kernel(const float* __restrict__ sums,
                            const float* __restrict__ cnt,
                            const float* __restrict__ Wlin,
                            const float* __restrict__ blin,
                            float* __restrict__ out) {
    int idx = blockIdx.x * blockDim.x + threadIdx.x;
    if (idx < NGR * OUTF) {
        int g = idx / OUTF;
        int o = idx % OUTF;
        float invc = 1.0f / fmaxf(cnt[g], 1.0f);
        float acc = blin[o];
        #pragma unroll 8
        for (int c = 0; c < HID; ++c)
            acc += sums[(long)g * HID + c] * invc * Wlin[c * OUTF + o];
        out[idx] = acc;
    }
}

// ---------------- gemm dispatch helper ----------------

template<int K, bool RELU>
static inline void launch_gemm(const float* H, const float* W, float* Out,
                               int nrows, hipStream_t stream) {
    const int full = nrows / 16;               // full 16-row blocks (branch-free path)
    const int rem  = nrows - full * 16;
    if (full > 0)
        gemm_wmma_kernel<K, RELU, false><<<full, 256, 0, stream>>>(H, W, Out, nrows, 0);
    if (rem > 0)
        gemm_wmma_kernel<K, RELU, true ><<<1, 256, 0, stream>>>(H, W, Out, nrows, full);
}

// ---------------- driver ----------------

extern "C" void kernel_launch(void* const* d_in, const int* in_sizes, int n_in,
                              void* d_out, int out_size, void* d_ws, size_t ws_size,
                              hipStream_t stream) {
    const float* x    = (const float*)d_in[0];
    const int*   ei   = (const int*)  d_in[1];   // [2, E]
    const int*   batch= (const int*)  d_in[2];   // [N]
    const float* W1   = (const float*)d_in[3];
    const float* b1   = (const float*)d_in[4];
    const float* W2   = (const float*)d_in[5];
    const float* b2   = (const float*)d_in[6];
    const float* W3   = (const float*)d_in[7];
    const float* b3   = (const float*)d_in[8];
    const float* W4   = (const float*)d_in[9];
    const float* b4   = (const float*)d_in[10];
    const float* Wlin = (const float*)d_in[11];
    const float* blin = (const float*)d_in[12];
    float* out = (float*)d_out;

    const int N = in_sizes[0] / FINF;
    const int E = in_sizes[1] / 2;
    const int* src = ei;
    const int* dst = ei + E;

    // workspace layout (floats)
    float* ws   = (float*)d_ws;
    float* dinv = ws;                          // N       (deg -> dinv in place)
    float* nrm  = dinv + N;                    // E
    float* bufA = nrm  + E;                    // N*HID   (h / agg output)
    float* bufB = bufA + (size_t)N * HID;      // N*HID   (hW scratch)
    float* sums = bufB + (size_t)N * HID;      // NGR*HID
    float* cnt  = sums + (size_t)NGR * HID;    // NGR

    const int B = 256;
    const long totNH  = (long)N * HID;
    const long totE32 = (long)E * 32;

    // normalization (once; reused by all 4 layers)
    init_deg_kernel<<<(N + B - 1) / B, B, 0, stream>>>(dinv, N);
    edge_deg_kernel<<<(E + B - 1) / B, B, 0, stream>>>(dst, dinv, E);
    dinv_kernel<<<(N + B - 1) / B, B, 0, stream>>>(dinv, N);
    norm_kernel<<<(E + B - 1) / B, B, 0, stream>>>(src, dst, dinv, nrm, E);

    // layer 1: x[N,7] @ W1 -> bufB ; agg -> bufA
    launch_gemm<FINF, false>(x, W1, bufB, N, stream);
    init_agg_kernel<<<(int)((totNH + B - 1) / B), B, 0, stream>>>(bufB, dinv, b1, bufA, totNH);
    edge_scatter_kernel<<<(int)((totE32 + B - 1) / B), B, 0, stream>>>(src, dst, nrm, bufB, bufA, totE32);

    // layers 2..4: relu fused into the LDS staging of A
    const float* Ws[3] = {W2, W3, W4};
    const float* bs[3] = {b2, b3, b4};
    for (int l = 0; l < 3; ++l) {
        launch_gemm<HID, true>(bufA, Ws[l], bufB, N, stream);
        init_agg_kernel<<<(int)((totNH + B - 1) / B), B, 0, stream>>>(bufB, dinv, bs[l], bufA, totNH);
        edge_scatter_kernel<<<(int)((totE32 + B - 1) / B), B, 0, stream>>>(src, dst, nrm, bufB, bufA, totE32);
    }

    // global mean pool + linear head
    pool_zero_kernel<<<(NGR * HID + B - 1) / B, B, 0, stream>>>(sums, cnt);
    pool_scatter_kernel<<<(int)((totNH + B - 1) / B), B, 0, stream>>>(bufA, batch, sums, cnt, totNH);
    head_kernel<<<(NGR * OUTF + B - 1) / B, B, 0, stream>>>(sums, cnt, Wlin, blin, out);
}